// OverlapAdd_60155311948306
// MI455X (gfx1250) — compile-verified
//
#include <hip/hip_runtime.h>

// Overlap-add, reformulated as a per-output gather (no atomics):
//   q = t/40, rem = t%40
//   out[t] = (q<=79 ? x[rem, q] : 0) + (q>=1 ? x[rem+40, q-1] : 0)
//            + (q==80 ? x[rem, 79] : 0)          per (b, f) plane
// Each block stages one (b,f) plane (80*81 = 6480 floats, 25.9 KB) into LDS
// with CDNA5 async global->LDS b128 transfers (ASYNCcnt path), then gathers
// from LDS and writes 3240 contiguous floats. Memory-bound: ~318 MB total ->
// ~14 us floor at 23.3 TB/s. WMMA not applicable (zero matmul FLOPs).

#define TWO_P        80
#define NCHUNK       81
#define PLANE_ELEMS  (TWO_P * NCHUNK)   // 6480 floats per (b,f) plane
#define NFRAMES      3240
#define BLOCK        256                // 8 wave32 waves

#if defined(__gfx1250__) && \
    __has_builtin(__builtin_amdgcn_global_load_async_to_lds_b128) && \
    __has_builtin(__builtin_amdgcn_s_wait_asynccnt)
#define USE_ASYNC_LDS 1
#else
#define USE_ASYNC_LDS 0
#endif

#if USE_ASYNC_LDS
typedef int v4i_vs __attribute__((vector_size(16)));           // b128 payload
typedef __attribute__((address_space(1))) v4i_vs* g_v4i_ptr;   // global
typedef __attribute__((address_space(3))) v4i_vs* l_v4i_ptr;   // LDS
#endif

__global__ __launch_bounds__(BLOCK)
void OverlapAdd_kernel(const float* __restrict__ x, float* __restrict__ out) {
    __shared__ float tile[PLANE_ELEMS];          // 25920 B LDS

    const int plane = blockIdx.x;                // b*512 + f
    const float* __restrict__ xin = x + (size_t)plane * PLANE_ELEMS;
    float* __restrict__ o = out + (size_t)plane * NFRAMES;
    const int tid = threadIdx.x;

    // ---- Stage plane into LDS, coalesced 16B transfers (6480/4 = 1620 b128s)
#if USE_ASYNC_LDS
    for (int v = tid; v < PLANE_ELEMS / 4; v += BLOCK) {
        __builtin_amdgcn_global_load_async_to_lds_b128(
            (g_v4i_ptr)(xin + 4 * v),
            (l_v4i_ptr)(tile + 4 * v),
            /*imm offset*/ 0, /*cpol*/ 0);
    }
    __builtin_amdgcn_s_wait_asynccnt(0);   // my async->LDS writes done
    __syncthreads();                       // everyone's writes visible
#else
    for (int v = tid; v < PLANE_ELEMS / 4; v += BLOCK) {
        ((float4*)tile)[v] = ((const float4*)xin)[v];
    }
    __syncthreads();
#endif

    // ---- Gather from LDS; stride-81 word accesses -> conflict-free banks
    for (int t = tid; t < NFRAMES; t += BLOCK) {
        const int q   = t / 40;
        const int rem = t - q * 40;
        float acc = (q <= NCHUNK - 2) ? tile[rem * NCHUNK + q] : 0.0f;
        if (q >= 1)
            acc += tile[(rem + TWO_P / 2) * NCHUNK + (q - 1)];
        if (q == NCHUNK - 1)                       // tail re-add of chunk 79
            acc += tile[rem * NCHUNK + (NCHUNK - 2)];
        o[t] = acc;                                // contiguous, coalesced
    }
}

extern "C" void kernel_launch(void* const* d_in, const int* in_sizes, int n_in,
                              void* d_out, int out_size, void* d_ws, size_t ws_size,
                              hipStream_t stream) {
    const float* x = (const float*)d_in[0];
    float* out = (float*)d_out;
    const int planes = in_sizes[0] / PLANE_ELEMS;  // 16*512 = 8192
    OverlapAdd_kernel<<<dim3(planes), dim3(BLOCK), 0, stream>>>(x, out);
}